// MVRM_33457795236543
// MI455X (gfx1250) — compile-verified
//
#include <hip/hip_runtime.h>
#include <hip/hip_bf16.h>

typedef float v2f __attribute__((ext_vector_type(2)));
typedef float v8f __attribute__((ext_vector_type(8)));

#define N 4096
#define TILES 256          // tiles per dim (4096/16)
#define WPB 8              // waves per block

// 3-level 1D Haar on 16 aligned samples as an orthonormal 16x16 matrix.
// rows 0-1: approx L3, 2-3: detail L3, 4-7: detail L2, 8-15: detail L1
__device__ __forceinline__ float T_elem(int i, int c) {
    const float s1 = 0.70710678118654752f;   // (1/sqrt2)
    const float s2 = 0.5f;                   // (1/sqrt2)^2
    const float s3 = 0.35355339059327379f;   // (1/sqrt2)^3
    if (i < 2)  return ((c >> 3) == i) ? s3 : 0.0f;
    if (i < 4)  return ((c >> 3) == (i - 2)) ? (((c >> 2) & 1) ? -s3 : s3) : 0.0f;
    if (i < 8)  return ((c >> 2) == (i - 4)) ? (((c >> 1) & 1) ? -s2 : s2) : 0.0f;
    return ((c >> 1) == (i - 8)) ? ((c & 1) ? -s1 : s1) : 0.0f;
}

// D = A(16x4 f32) * B(4x16 f32) + C  — chained along K.
__device__ __forceinline__ v8f wmma4(v2f a, v2f b, v8f c) {
    return __builtin_amdgcn_wmma_f32_16x16x4_f32(
        /*neg_a=*/false, a, /*neg_b=*/false, b,
        /*c_mod=*/(short)0, c, /*reuse_a=*/false, /*reuse_b=*/false);
}

// ---------------------------------------------------------------------------
// Forward: per 16x16 tile compute Z = (T X T^t)^t = T * (T*X)^t via 2 WMMA
// chains + an LDS transpose. Store Z tile-wise into Ymid (coalesced) and the
// exact Mallat HH1/HH2/HH3 coefficients into dense subband images.
// ---------------------------------------------------------------------------
__global__ __launch_bounds__(WPB * 32)
void haar_fwd_wmma(const float* __restrict__ x, float* __restrict__ Ymid,
                   float* __restrict__ hh1, float* __restrict__ hh2,
                   float* __restrict__ hh3) {
    __shared__ float lds[WPB * 320];              // 16 rows * stride 20 per wave
    const int wid  = threadIdx.x >> 5;
    const int lane = threadIdx.x & 31;
    const int tile = blockIdx.x * WPB + wid;
    const int ty = tile >> 8, tx = tile & (TILES - 1);
    const int n = lane & 15, h = lane >> 4;

    // A operand: constant T in 16x4-chunk A layout (lanes 0-15: K 0,1; 16-31: K 2,3)
    v2f a[4];
#pragma unroll
    for (int kc = 0; kc < 4; ++kc) {
        a[kc].x = T_elem(n, 4 * kc + 2 * h);
        a[kc].y = T_elem(n, 4 * kc + 2 * h + 1);
    }

    // Chain 1: M = T * X  (X loaded coalesced in B layout)
    const float* xp = x + (size_t)(16 * ty) * N + 16 * tx + n;
    v8f acc = {};
#pragma unroll
    for (int kc = 0; kc < 4; ++kc) {
        v2f b;
        b.x = xp[(4 * kc + 2 * h) * N];
        b.y = xp[(4 * kc + 2 * h + 1) * N];
        acc = wmma4(a[kc], b, acc);
    }

    // Transpose M through LDS (store M^t row-major, padded stride 20 words)
    float* lp = &lds[wid * 320 + n * 20 + 8 * h];
#pragma unroll
    for (int r = 0; r < 8; ++r) lp[r] = acc[r];   // same-wave DS ops are in-order

    // Chain 2: Z = T * M^t  ( = Y^t )
    const float* lq = &lds[wid * 320];
    v8f z = {};
#pragma unroll
    for (int kc = 0; kc < 4; ++kc) {
        v2f b;
        b.x = lq[(4 * kc + 2 * h) * 20 + n];
        b.y = lq[(4 * kc + 2 * h + 1) * 20 + n];
        z = wmma4(a[kc], b, z);
    }
    // C layout: z[r] = Z[r + 8h][n]

    // Full tile -> Ymid (coalesced column-across-lanes stores)
    float* yp = Ymid + (size_t)(16 * ty) * N + 16 * tx + n;
#pragma unroll
    for (int r = 0; r < 8; ++r) yp[(r + 8 * h) * N] = z[r];

    // HH scatter: Z[i][j] = Y[j][i]; HH-L = {i,j in [bs,2bs)}, bs = 8/4/2
    if (h == 1 && n >= 8) {                       // HH1 (8x8 block, row-contiguous)
        float* p = hh1 + (size_t)(8 * ty + n - 8) * 2048 + 8 * tx;
#pragma unroll
        for (int r = 0; r < 8; ++r) p[r] = z[r];
    }
    if (h == 0 && n >= 4 && n < 8) {              // HH2 (4x4 block)
        float* p = hh2 + (size_t)(4 * ty + n - 4) * 1024 + 4 * tx;
#pragma unroll
        for (int r = 4; r < 8; ++r) p[r - 4] = z[r];
    }
    if (h == 0 && (n == 2 || n == 3)) {           // HH3 (2x2 block)
        float* p = hh3 + (size_t)(2 * ty + n - 2) * 512 + 2 * tx;
        p[0] = z[2];
        p[1] = z[3];
    }
}

// ---------------------------------------------------------------------------
// SAME cross-correlation (XLA conv semantics) on an HH image; result written
// directly into the matching Z-layout slots of Ymid.
// ---------------------------------------------------------------------------
template <int K, int BS, int S>
__global__ __launch_bounds__(256)
void conv_hh(const float* __restrict__ in, const float* __restrict__ w,
             const float* __restrict__ bias, float* __restrict__ Ymid) {
    const int idx = blockIdx.x * blockDim.x + threadIdx.x;
    const int a = idx / S, b = idx % S;
    constexpr int R = K / 2;
    float acc = bias[0];
#pragma unroll
    for (int ky = 0; ky < K; ++ky) {
#pragma unroll
        for (int kx = 0; kx < K; ++kx) {
            const int ia = a + ky - R, ib = b + kx - R;
            if (ia >= 0 && ia < S && ib >= 0 && ib < S)
                acc += w[ky * K + kx] * in[(size_t)ia * S + ib];
        }
    }
    // hh(a,b) lives at Z[BS + (b%BS)][BS + (a%BS)] of tile (a/BS, b/BS)
    Ymid[(size_t)(16 * (a / BS) + BS + (b % BS)) * N +
         (16 * (b / BS) + BS + (a % BS))] = acc;
}

// ---------------------------------------------------------------------------
// Inverse: X = T^t * (T^t * Z)^t  (since Z = Y^t and X = T^t Y T)
// ---------------------------------------------------------------------------
__global__ __launch_bounds__(WPB * 32)
void haar_inv_wmma(const float* __restrict__ Ymid, float* __restrict__ out) {
    __shared__ float lds[WPB * 320];
    const int wid  = threadIdx.x >> 5;
    const int lane = threadIdx.x & 31;
    const int tile = blockIdx.x * WPB + wid;
    const int ty = tile >> 8, tx = tile & (TILES - 1);
    const int n = lane & 15, h = lane >> 4;

    // A operand: constant T^t chunks
    v2f a[4];
#pragma unroll
    for (int kc = 0; kc < 4; ++kc) {
        a[kc].x = T_elem(4 * kc + 2 * h, n);
        a[kc].y = T_elem(4 * kc + 2 * h + 1, n);
    }

    // Chain 1: P = T^t * Z  (Z loaded coalesced from Ymid)
    const float* yp = Ymid + (size_t)(16 * ty) * N + 16 * tx + n;
    v8f acc = {};
#pragma unroll
    for (int kc = 0; kc < 4; ++kc) {
        v2f b;
        b.x = yp[(4 * kc + 2 * h) * N];
        b.y = yp[(4 * kc + 2 * h + 1) * N];
        acc = wmma4(a[kc], b, acc);
    }

    // Transpose P through LDS
    float* lp = &lds[wid * 320 + n * 20 + 8 * h];
#pragma unroll
    for (int r = 0; r < 8; ++r) lp[r] = acc[r];

    // Chain 2: Q = T^t * P^t = T^t Y T = X
    const float* lq = &lds[wid * 320];
    v8f q = {};
#pragma unroll
    for (int kc = 0; kc < 4; ++kc) {
        v2f b;
        b.x = lq[(4 * kc + 2 * h) * 20 + n];
        b.y = lq[(4 * kc + 2 * h + 1) * 20 + n];
        q = wmma4(a[kc], b, q);
    }

    float* op = out + (size_t)(16 * ty) * N + 16 * tx + n;
#pragma unroll
    for (int r = 0; r < 8; ++r) op[(r + 8 * h) * N] = q[r];
}

extern "C" void kernel_launch(void* const* d_in, const int* in_sizes, int n_in,
                              void* d_out, int out_size, void* d_ws, size_t ws_size,
                              hipStream_t stream) {
    const float* x  = (const float*)d_in[0];
    const float* w3 = (const float*)d_in[1];
    const float* b3 = (const float*)d_in[2];
    const float* w5 = (const float*)d_in[3];
    const float* b5 = (const float*)d_in[4];
    const float* w7 = (const float*)d_in[5];
    const float* b7 = (const float*)d_in[6];
    float* out = (float*)d_out;

    char* ws = (char*)d_ws;
    float* Ymid = (float*)(ws);                                  // 64 MB
    float* hh1  = (float*)(ws + (size_t)64 * 1024 * 1024);       // 16 MB (2048^2)
    float* hh2  = (float*)(ws + (size_t)80 * 1024 * 1024);       //  4 MB (1024^2)
    float* hh3  = (float*)(ws + (size_t)84 * 1024 * 1024);       //  1 MB (512^2)

    const int tileBlocks = (TILES * TILES) / WPB;                // 8192

    haar_fwd_wmma<<<tileBlocks, WPB * 32, 0, stream>>>(x, Ymid, hh1, hh2, hh3);

    conv_hh<7, 8, 2048><<<(2048 * 2048) / 256, 256, 0, stream>>>(hh1, w7, b7, Ymid);
    conv_hh<5, 4, 1024><<<(1024 * 1024) / 256, 256, 0, stream>>>(hh2, w5, b5, Ymid);
    conv_hh<3, 2, 512 ><<<(512  * 512 ) / 256, 256, 0, stream>>>(hh3, w3, b3, Ymid);

    haar_inv_wmma<<<tileBlocks, WPB * 32, 0, stream>>>(Ymid, out);
}